// LRU_56135222559197
// MI455X (gfx1250) — compile-verified
//
#include <hip/hip_runtime.h>
#include <cstddef>
#include <cstdint>

// ---------------------------------------------------------------------------
// LRU forward for MI455X (gfx1250, wave32, WMMA).
//   B_SZ=8, SEQ=8192, N=256, R = B_SZ*SEQ = 65536
// Pipeline:
//   K0a precompute weights: Brn=B*exp(gamma), Crn=Re(C), Cin=-Im(C)
//   K0b per-channel lambda, lambda^128, and pow table lambda^(j+1), j<128
//   K1  Bx(re,im) = X @ Brnᵀ        (fp32 WMMA 16x16x4, LDS-staged tiles)
//   K2a chunked local scan (64 chunks of 128) in-place on Bx planes
//   K2b tiny cross-chunk carry scan using lambda^128
//   K3  out = (Sr+corr_r) @ Crᵀ + (Si+corr_i) @ (-Ci)ᵀ, where the chunk-carry
//       correction corr = lambda^(j+1)*carry is fused into LDS staging,
//       eliminating the 256 MB read+write pass a separate apply kernel needs.
// Workload is HBM-bound (~0.65 GB traffic after fusion @ 23.3 TB/s ≈ 28 us);
// fp32 WMMA keeps reference precision at negligible compute cost (34 GFLOP).
// ---------------------------------------------------------------------------

typedef float v2f __attribute__((ext_vector_type(2)));
typedef float v8f __attribute__((ext_vector_type(8)));

static constexpr int N_DIM  = 256;
static constexpr int SEQ    = 8192;
static constexpr int B_SZ   = 8;
static constexpr int R_ROWS = B_SZ * SEQ;        // 65536
static constexpr int CHUNK  = 128;               // scan chunk length (2^7)
static constexpr int NCHUNK = SEQ / CHUNK;       // 64
static constexpr int BM     = 64;                // GEMM block tile rows
static constexpr int BN     = 64;                // GEMM block tile cols
static constexpr int LDK    = 260;               // padded LDS row stride (dwords)

// ---------------------------------------------------------------------------
// K0a: elementwise precompute over the 256x256 weight matrices.
// ---------------------------------------------------------------------------
__global__ __launch_bounds__(256) void k_precompute(
    const float* __restrict__ gamma_log,
    const float2* __restrict__ Bc, const float2* __restrict__ Cc,
    float* __restrict__ Brn_re, float* __restrict__ Brn_im,
    float* __restrict__ Crn_re, float* __restrict__ Crn_imn)
{
    const int idx = blockIdx.x * 256 + threadIdx.x;      // 65536 total
    const int m = idx >> 8;
    const float eg = __expf(gamma_log[m]);
    const float2 b = Bc[idx];
    Brn_re[idx] = b.x * eg;
    Brn_im[idx] = b.y * eg;
    const float2 c = Cc[idx];
    Crn_re[idx]  = c.x;
    Crn_imn[idx] = -c.y;                                 // fold the minus sign
}

// ---------------------------------------------------------------------------
// K0b: per-channel lambda, lambda^CHUNK (7 complex squarings) and the
// lambda^(j+1) power table for the fused carry correction. One block.
// ---------------------------------------------------------------------------
__global__ __launch_bounds__(256) void k_lambda(
    const float* __restrict__ nu_log, const float* __restrict__ theta_log,
    float* __restrict__ lam_re, float* __restrict__ lam_im,
    float* __restrict__ lamT_re, float* __restrict__ lamT_im,
    float* __restrict__ pow_re, float* __restrict__ pow_im)
{
    const int n = threadIdx.x;
    const float lmod = __expf(-__expf(nu_log[n]));
    const float th   = __expf(theta_log[n]);
    const float lr = lmod * __cosf(th);
    const float li = lmod * __sinf(th);
    lam_re[n] = lr;
    lam_im[n] = li;

    float tr = lr, ti = li;                              // lambda^(2^7)
    #pragma unroll
    for (int i = 0; i < 7; ++i) {
        const float nr = tr * tr - ti * ti;
        const float ni = 2.0f * tr * ti;
        tr = nr; ti = ni;
    }
    lamT_re[n] = tr;
    lamT_im[n] = ti;

    float fr = lr, fi = li;                              // lambda^(j+1)
    for (int j = 0; j < CHUNK; ++j) {
        pow_re[j * N_DIM + n] = fr;
        pow_im[j * N_DIM + n] = fi;
        const float nr = fr * lr - fi * li;
        const float ni = fr * li + fi * lr;
        fr = nr; fi = ni;
    }
}

// ---------------------------------------------------------------------------
// K1: Or/Oi[r, n] = sum_k X[r, k] * Wr/Wi[n, k]   (K = 256, fully LDS-staged)
// Block = 256 threads = 8 waves: wave covers a 16(M) x 32(N) patch, two
// 16x16 WMMA n-tiles, for both re and im weight planes.
// LDS = (BM + 2*BN) * LDK * 4 = 195 KB (fits CDNA5's 320 KB WGP LDS).
// LDS row stride 260 dwords -> ds_load_b64 fragments are bank-conflict-free.
// ---------------------------------------------------------------------------
__global__ __launch_bounds__(256) void k_gemm_b(
    const float* __restrict__ X,
    const float* __restrict__ Wr, const float* __restrict__ Wi,
    float* __restrict__ Or, float* __restrict__ Oi)
{
    extern __shared__ float lds[];
    float* Xs  = lds;                    // BM x LDK
    float* Wrs = Xs  + BM * LDK;         // BN x LDK
    float* Wis = Wrs + BN * LDK;         // BN x LDK

    const int tid = threadIdx.x;
    const int r0  = blockIdx.x * BM;
    const int n0  = blockIdx.y * BN;

    const float4* Xg  = (const float4*)(X  + (size_t)r0 * N_DIM);
    const float4* Wrg = (const float4*)(Wr + (size_t)n0 * N_DIM);
    const float4* Wig = (const float4*)(Wi + (size_t)n0 * N_DIM);
    #pragma unroll
    for (int i = 0; i < (BM * N_DIM / 4) / 256; ++i) {   // 16 iters
        const int f   = i * 256 + tid;
        const int row = f >> 6;                          // 64 float4 per row
        const int c4  = (f & 63) << 2;
        *(float4*)&Xs [row * LDK + c4] = Xg [f];
        *(float4*)&Wrs[row * LDK + c4] = Wrg[f];
        *(float4*)&Wis[row * LDK + c4] = Wig[f];
    }
    __syncthreads();

    const int wave  = tid >> 5;
    const int lane  = tid & 31;
    const int lm    = lane & 15;
    const int khalf = (lane >> 4) << 1;        // lanes 16-31 hold K+2,K+3
    const int wm    = wave & 3;                // 4 M subtiles of 16 rows
    const int wn    = wave >> 2;               // 2 N groups of 32 cols

    const float* arow = &Xs [(wm * 16 + lm) * LDK + khalf];
    const float* br0  = &Wrs[(wn * 32 +      lm) * LDK + khalf];
    const float* br1  = &Wrs[(wn * 32 + 16 + lm) * LDK + khalf];
    const float* bi0  = &Wis[(wn * 32 +      lm) * LDK + khalf];
    const float* bi1  = &Wis[(wn * 32 + 16 + lm) * LDK + khalf];

    v8f accR0 = {}, accR1 = {}, accI0 = {}, accI1 = {};
    for (int k = 0; k < N_DIM; k += 4) {
        const v2f a  = *(const v2f*)&arow[k];
        const v2f b0 = *(const v2f*)&br0[k];
        const v2f b1 = *(const v2f*)&br1[k];
        const v2f c0 = *(const v2f*)&bi0[k];
        const v2f c1 = *(const v2f*)&bi1[k];
        accR0 = __builtin_amdgcn_wmma_f32_16x16x4_f32(false, a, false, b0, (short)0, accR0, false, false);
        accR1 = __builtin_amdgcn_wmma_f32_16x16x4_f32(false, a, false, b1, (short)0, accR1, false, false);
        accI0 = __builtin_amdgcn_wmma_f32_16x16x4_f32(false, a, false, c0, (short)0, accI0, false, false);
        accI1 = __builtin_amdgcn_wmma_f32_16x16x4_f32(false, a, false, c1, (short)0, accI1, false, false);
    }

    // D layout: VGPR j holds (M = j + 8*(lane>=16), N = lane&15)
    const int mrow = r0 + wm * 16 + ((lane >> 4) << 3);
    const int col  = n0 + wn * 32 + lm;
    #pragma unroll
    for (int j = 0; j < 8; ++j) {
        const size_t rb = (size_t)(mrow + j) * N_DIM;
        Or[rb + col]      = accR0[j];
        Or[rb + col + 16] = accR1[j];
        Oi[rb + col]      = accI0[j];
        Oi[rb + col + 16] = accI1[j];
    }
}

// ---------------------------------------------------------------------------
// K2a: local scan inside each chunk (in place), emit chunk aggregate.
// Grid = (NCHUNK, B_SZ), 256 threads = one channel each; fully coalesced.
// ---------------------------------------------------------------------------
__global__ __launch_bounds__(256) void k_scan_local(
    float* __restrict__ Sr, float* __restrict__ Si,
    const float* __restrict__ lam_re, const float* __restrict__ lam_im,
    float* __restrict__ agg_re, float* __restrict__ agg_im)
{
    const int n = threadIdx.x;
    const int c = blockIdx.x;
    const int b = blockIdx.y;
    const float lr = lam_re[n], li = lam_im[n];
    size_t p = ((size_t)(b * SEQ + c * CHUNK)) * N_DIM + n;
    float hr = 0.0f, hi = 0.0f;
    for (int t = 0; t < CHUNK; ++t, p += N_DIM) {
        const float ur = Sr[p], ui = Si[p];
        const float nr = lr * hr - li * hi + ur;
        const float ni = lr * hi + li * hr + ui;
        hr = nr; hi = ni;
        Sr[p] = hr; Si[p] = hi;
    }
    const size_t ap = ((size_t)(b * NCHUNK + c)) * N_DIM + n;
    agg_re[ap] = hr;
    agg_im[ap] = hi;
}

// ---------------------------------------------------------------------------
// K2b: exclusive scan of chunk aggregates across the 64 chunks (per b,n),
// with multiplier lambda^CHUNK. 2048 threads, trivial cost.
// ---------------------------------------------------------------------------
__global__ __launch_bounds__(256) void k_scan_carry(
    const float* __restrict__ agg_re, const float* __restrict__ agg_im,
    const float* __restrict__ lamT_re, const float* __restrict__ lamT_im,
    float* __restrict__ car_re, float* __restrict__ car_im)
{
    const int n = threadIdx.x;
    const int b = blockIdx.x;
    const float tr = lamT_re[n], ti = lamT_im[n];
    float cr = 0.0f, ci = 0.0f;
    for (int c = 0; c < NCHUNK; ++c) {
        const size_t p = ((size_t)(b * NCHUNK + c)) * N_DIM + n;
        car_re[p] = cr;                      // exclusive carry into chunk c
        car_im[p] = ci;
        const float ar = agg_re[p], ai = agg_im[p];
        const float nr = tr * cr - ti * ci + ar;
        const float ni = tr * ci + ti * cr + ai;
        cr = nr; ci = ni;
    }
}

// ---------------------------------------------------------------------------
// K3: out[r, n] = sum_k S'r[r,k]*Cr[n,k] + S'i[r,k]*Cin[n,k]  (Cin = -Im C)
// where S' = localstate + lambda^(j+1)*carry is formed on the fly while
// staging the A tiles into LDS (pow/carry tables are L2-resident).
// LDS = 4 * 64 * 260 * 4 = 260 KB dynamic.
// ---------------------------------------------------------------------------
__global__ __launch_bounds__(256) void k_gemm_out(
    const float* __restrict__ Sr, const float* __restrict__ Si,
    const float* __restrict__ Cr, const float* __restrict__ Cin,
    const float* __restrict__ pow_re, const float* __restrict__ pow_im,
    const float* __restrict__ car_re, const float* __restrict__ car_im,
    float* __restrict__ Out)
{
    extern __shared__ float lds[];
    float* Srs = lds;                    // BM x LDK
    float* Sis = Srs + BM * LDK;         // BM x LDK
    float* Crs = Sis + BM * LDK;         // BN x LDK
    float* Cis = Crs + BN * LDK;         // BN x LDK

    const int tid = threadIdx.x;
    const int r0  = blockIdx.x * BM;
    const int n0  = blockIdx.y * BN;

    const float4* Sg  = (const float4*)(Sr  + (size_t)r0 * N_DIM);
    const float4* Ig  = (const float4*)(Si  + (size_t)r0 * N_DIM);
    const float4* Crg = (const float4*)(Cr  + (size_t)n0 * N_DIM);
    const float4* Cig = (const float4*)(Cin + (size_t)n0 * N_DIM);
    #pragma unroll
    for (int i = 0; i < (BM * N_DIM / 4) / 256; ++i) {
        const int f   = i * 256 + tid;
        const int row = f >> 6;
        const int c4  = (f & 63) << 2;

        // fused chunk-carry correction: S' = S + lambda^(j+1) * carry
        const int gr = r0 + row;                 // global state row
        const int b  = gr >> 13;                 // / SEQ
        const int t  = gr & (SEQ - 1);
        const int ch = t >> 7;                   // / CHUNK
        const int j  = t & (CHUNK - 1);
        float4 sr4 = Sg[f];
        float4 si4 = Ig[f];
        const float4 pr = *(const float4*)&pow_re[j * N_DIM + c4];
        const float4 pi = *(const float4*)&pow_im[j * N_DIM + c4];
        const size_t cb = ((size_t)(b * NCHUNK + ch)) * N_DIM + c4;
        const float4 cr = *(const float4*)&car_re[cb];
        const float4 ci = *(const float4*)&car_im[cb];
        sr4.x += pr.x * cr.x - pi.x * ci.x;  si4.x += pr.x * ci.x + pi.x * cr.x;
        sr4.y += pr.y * cr.y - pi.y * ci.y;  si4.y += pr.y * ci.y + pi.y * cr.y;
        sr4.z += pr.z * cr.z - pi.z * ci.z;  si4.z += pr.z * ci.z + pi.z * cr.z;
        sr4.w += pr.w * cr.w - pi.w * ci.w;  si4.w += pr.w * ci.w + pi.w * cr.w;

        *(float4*)&Srs[row * LDK + c4] = sr4;
        *(float4*)&Sis[row * LDK + c4] = si4;
        *(float4*)&Crs[row * LDK + c4] = Crg[f];
        *(float4*)&Cis[row * LDK + c4] = Cig[f];
    }
    __syncthreads();

    const int wave  = tid >> 5;
    const int lane  = tid & 31;
    const int lm    = lane & 15;
    const int khalf = (lane >> 4) << 1;
    const int wm    = wave & 3;
    const int wn    = wave >> 2;

    const float* ar  = &Srs[(wm * 16 + lm) * LDK + khalf];
    const float* ai  = &Sis[(wm * 16 + lm) * LDK + khalf];
    const float* br0 = &Crs[(wn * 32 +      lm) * LDK + khalf];
    const float* br1 = &Crs[(wn * 32 + 16 + lm) * LDK + khalf];
    const float* bi0 = &Cis[(wn * 32 +      lm) * LDK + khalf];
    const float* bi1 = &Cis[(wn * 32 + 16 + lm) * LDK + khalf];

    v8f acc0 = {}, acc1 = {};
    for (int k = 0; k < N_DIM; k += 4) {
        const v2f vr  = *(const v2f*)&ar[k];
        const v2f vi  = *(const v2f*)&ai[k];
        const v2f wr0 = *(const v2f*)&br0[k];
        const v2f wr1 = *(const v2f*)&br1[k];
        const v2f wi0 = *(const v2f*)&bi0[k];
        const v2f wi1 = *(const v2f*)&bi1[k];
        acc0 = __builtin_amdgcn_wmma_f32_16x16x4_f32(false, vr, false, wr0, (short)0, acc0, false, false);
        acc0 = __builtin_amdgcn_wmma_f32_16x16x4_f32(false, vi, false, wi0, (short)0, acc0, false, false);
        acc1 = __builtin_amdgcn_wmma_f32_16x16x4_f32(false, vr, false, wr1, (short)0, acc1, false, false);
        acc1 = __builtin_amdgcn_wmma_f32_16x16x4_f32(false, vi, false, wi1, (short)0, acc1, false, false);
    }

    const int mrow = r0 + wm * 16 + ((lane >> 4) << 3);
    const int col  = n0 + wn * 32 + lm;
    #pragma unroll
    for (int j = 0; j < 8; ++j) {
        const size_t rb = (size_t)(mrow + j) * N_DIM;
        Out[rb + col]      = acc0[j];
        Out[rb + col + 16] = acc1[j];
    }
}

// ---------------------------------------------------------------------------
// Host-side launch. Workspace layout (all fp32 planes, ~138 MB total):
//   Brn_re | Brn_im | Crn_re | Crn_imn            4 * 256 KB
//   lam_re | lam_im | lamT_re | lamT_im           4 * 1 KB
//   pow_re | pow_im                               2 * 128 KB
//   Bx_re | Bx_im (also hold states in place)     2 * 64 MB
//   agg_re | agg_im | car_re | car_im             4 * 512 KB
// ---------------------------------------------------------------------------
extern "C" void kernel_launch(void* const* d_in, const int* in_sizes, int n_in,
                              void* d_out, int out_size, void* d_ws, size_t ws_size,
                              hipStream_t stream) {
    const float*  x         = (const float*)d_in[0];   // [8, 8192, 256]
    const float*  nu_log    = (const float*)d_in[1];   // [256]
    const float*  theta_log = (const float*)d_in[2];   // [256]
    const float*  gamma_log = (const float*)d_in[3];   // [256]
    const float2* Bc        = (const float2*)d_in[4];  // [256, 256] complex64
    const float2* Cc        = (const float2*)d_in[5];  // [256, 256] complex64
    float*        out       = (float*)d_out;           // [8, 8192, 256]

    char* ws = (char*)d_ws;
    const size_t MAT   = (size_t)N_DIM * N_DIM * sizeof(float);        // 256 KB
    const size_t VECN  = (size_t)N_DIM * sizeof(float);                // 1 KB
    const size_t POW   = (size_t)CHUNK * N_DIM * sizeof(float);        // 128 KB
    const size_t PLANE = (size_t)R_ROWS * N_DIM * sizeof(float);       // 64 MB
    const size_t AGG   = (size_t)B_SZ * NCHUNK * N_DIM * sizeof(float);// 512 KB

    float* Brn_re  = (float*)(ws);                 ws += MAT;
    float* Brn_im  = (float*)(ws);                 ws += MAT;
    float* Crn_re  = (float*)(ws);                 ws += MAT;
    float* Crn_imn = (float*)(ws);                 ws += MAT;
    float* lam_re  = (float*)(ws);                 ws += VECN;
    float* lam_im  = (float*)(ws);                 ws += VECN;
    float* lamT_re = (float*)(ws);                 ws += VECN;
    float* lamT_im = (float*)(ws);                 ws += VECN;
    float* pow_re  = (float*)(ws);                 ws += POW;
    float* pow_im  = (float*)(ws);                 ws += POW;
    float* Bx_re   = (float*)(ws);                 ws += PLANE;
    float* Bx_im   = (float*)(ws);                 ws += PLANE;
    float* agg_re  = (float*)(ws);                 ws += AGG;
    float* agg_im  = (float*)(ws);                 ws += AGG;
    float* car_re  = (float*)(ws);                 ws += AGG;
    float* car_im  = (float*)(ws);                 ws += AGG;

    // K0a/K0b: weight and lambda precompute
    k_precompute<<<N_DIM, 256, 0, stream>>>(
        gamma_log, Bc, Cc, Brn_re, Brn_im, Crn_re, Crn_imn);
    k_lambda<<<1, 256, 0, stream>>>(
        nu_log, theta_log, lam_re, lam_im, lamT_re, lamT_im, pow_re, pow_im);

    // K1: Bx = X @ Brnᵀ (complex via two real planes)
    {
        dim3 grid(R_ROWS / BM, N_DIM / BN);
        const size_t lbytes = (size_t)(BM + 2 * BN) * LDK * sizeof(float); // 195 KB
        k_gemm_b<<<grid, 256, lbytes, stream>>>(x, Brn_re, Brn_im, Bx_re, Bx_im);
    }

    // K2: chunked associative scan (local pass in place + tiny carry scan)
    {
        dim3 grid(NCHUNK, B_SZ);
        k_scan_local<<<grid, 256, 0, stream>>>(Bx_re, Bx_im, lam_re, lam_im,
                                               agg_re, agg_im);
        k_scan_carry<<<B_SZ, 256, 0, stream>>>(agg_re, agg_im, lamT_re, lamT_im,
                                               car_re, car_im);
    }

    // K3: out = Re(states @ Cᵀ) with carry correction fused into staging
    {
        dim3 grid(R_ROWS / BM, N_DIM / BN);
        const size_t lbytes = (size_t)(2 * BM + 2 * BN) * LDK * sizeof(float); // 260 KB
        k_gemm_out<<<grid, 256, lbytes, stream>>>(Bx_re, Bx_im, Crn_re, Crn_imn,
                                                  pow_re, pow_im, car_re, car_im, out);
    }
}